// CPDecompPytorch3D_25254407701270
// MI455X (gfx1250) — compile-verified
//
#include <hip/hip_runtime.h>

// CP-decomposition loss for MI455X (gfx1250, wave32).
// Pass 1: gather-reduce, one observation per lane, grid-stride; block partials -> d_ws.
// Pass 2: deterministic single-block reduction of partials -> d_out[0].
// Wave-level reduction uses V_WMMA_F32_16X16X4_F32 as a one-shot reduction tree.

typedef __attribute__((ext_vector_type(2))) float v2f;
typedef __attribute__((ext_vector_type(8))) float v8f;

#define RANK 10
#define NBLOCKS 4096
#define BLOCK 256   // 8 waves of 32

// Full wave32 sum via one WMMA:
//   A (16x4 f32): VGPR0 = lane value, VGPR1 = 0.
//     lanes 0-15  -> A[lane,   0] = val
//     lanes 16-31 -> A[lane-16,2] = val      (each lane's value appears exactly once)
//   B (4x16) = all ones  =>  D[m,n] = val_m + val_{m+16}  for every n.
//   C/D layout: VGPR g holds M=g (lanes 0-15) and M=g+8 (lanes 16-31), N=lane%16.
//   Sum of the 8 C VGPRs per lane = half-sum; shfl_xor(16) completes the wave sum.
__device__ __forceinline__ float wave_sum_wmma(float v) {
    v2f a; a.x = v;    a.y = 0.0f;
    v2f b; b.x = 1.0f; b.y = 1.0f;
    v8f c = {};
    c = __builtin_amdgcn_wmma_f32_16x16x4_f32(
        /*neg_a=*/false, a, /*neg_b=*/false, b,
        /*c_mod=*/(short)0, c, /*reuse_a=*/false, /*reuse_b=*/false);
    float s = (c[0] + c[1]) + (c[2] + c[3]) + (c[4] + c[5]) + (c[6] + c[7]);
    s += __shfl_xor(s, 16);
    return s;  // every lane holds the full wave32 sum
}

__global__ __launch_bounds__(BLOCK) void cp_partial_kernel(
    const int* __restrict__ idx,   // [N,3] int32
    const float* __restrict__ y,   // [N]
    const float* __restrict__ U0,  // [D0,RANK]
    const float* __restrict__ U1,
    const float* __restrict__ U2,
    float* __restrict__ partials,  // [NBLOCKS]
    int N)
{
    float acc = 0.0f;
    const int stride = gridDim.x * blockDim.x;
    for (int n = blockIdx.x * blockDim.x + threadIdx.x; n < N; n += stride) {
        const size_t base = 3u * (size_t)n;
        // Streamed once per pass: keep out of L2 with non-temporal hints so the
        // 12MB factor tables stay resident (192MB L2 holds them trivially).
        const int i = __builtin_nontemporal_load(idx + base + 0);
        const int j = __builtin_nontemporal_load(idx + base + 1);
        const int k = __builtin_nontemporal_load(idx + base + 2);
        const float yn = __builtin_nontemporal_load(y + n);

        // Rows are 40B => 8-byte aligned: five b64 loads each (L2-resident gathers).
        const float2* __restrict__ ra = reinterpret_cast<const float2*>(U0 + (size_t)i * RANK);
        const float2* __restrict__ rb = reinterpret_cast<const float2*>(U1 + (size_t)j * RANK);
        const float2* __restrict__ rc = reinterpret_cast<const float2*>(U2 + (size_t)k * RANK);

        float pred = 0.0f;
#pragma unroll
        for (int r = 0; r < RANK / 2; ++r) {
            const float2 av = ra[r];
            const float2 bv = rb[r];
            const float2 cv = rc[r];
            pred = fmaf(av.x * bv.x, cv.x, pred);
            pred = fmaf(av.y * bv.y, cv.y, pred);
        }
        const float e = pred - yn;
        acc = fmaf(e, e, acc);
    }

    // All lanes reconverged here: EXEC all-1s, WMMA legal.
    const float wsum = wave_sum_wmma(acc);

    __shared__ float smem[BLOCK / 32];
    if ((threadIdx.x & 31) == 0) smem[threadIdx.x >> 5] = wsum;
    __syncthreads();
    if (threadIdx.x == 0) {
        float bs = 0.0f;
#pragma unroll
        for (int w = 0; w < BLOCK / 32; ++w) bs += smem[w];
        partials[blockIdx.x] = bs;  // fixed order -> deterministic
    }
}

__global__ __launch_bounds__(BLOCK) void cp_final_kernel(
    const float* __restrict__ partials, int nparts, float* __restrict__ out)
{
    float acc = 0.0f;
    for (int p = threadIdx.x; p < nparts; p += BLOCK) acc += partials[p];

    const float wsum = wave_sum_wmma(acc);

    __shared__ float smem[BLOCK / 32];
    if ((threadIdx.x & 31) == 0) smem[threadIdx.x >> 5] = wsum;
    __syncthreads();
    if (threadIdx.x == 0) {
        float bs = 0.0f;
#pragma unroll
        for (int w = 0; w < BLOCK / 32; ++w) bs += smem[w];
        out[0] = bs;  // overwrite (d_out is poisoned, never pre-zeroed)
    }
}

extern "C" void kernel_launch(void* const* d_in, const int* in_sizes, int n_in,
                              void* d_out, int out_size, void* d_ws, size_t ws_size,
                              hipStream_t stream) {
    const int*   idx = (const int*)  d_in[0];   // indices [N,3] (int32 per harness contract)
    const float* y   = (const float*)d_in[1];   // y [N]
    const float* U0  = (const float*)d_in[2];
    const float* U1  = (const float*)d_in[3];
    const float* U2  = (const float*)d_in[4];
    float* out       = (float*)d_out;
    float* partials  = (float*)d_ws;            // NBLOCKS floats = 16KB scratch

    const int N = in_sizes[1];                  // element count of y

    cp_partial_kernel<<<NBLOCKS, BLOCK, 0, stream>>>(idx, y, U0, U1, U2, partials, N);
    cp_final_kernel<<<1, BLOCK, 0, stream>>>(partials, NBLOCKS, out);
}